// MultiHeadSemGConv_39840116638374
// MI455X (gfx1250) — compile-verified
//
#include <hip/hip_runtime.h>
#include <math.h>

// Problem constants (from reference)
#define NB    98      // landmarks N
#define KNZ   10      // neighbors per row
#define NH    4       // heads
#define CIN   256     // input channels
#define HC    64      // head channels
#define COUT  256     // NH*HC output channels
#define MPAD  112     // 7 * 16 row tiles covering 98 rows
#define XS    260     // padded LDS row stride for x (bank-conflict-free A frags)
#define H1S   260     // padded LDS row stride for h1

typedef __attribute__((ext_vector_type(2))) float        v2f;
typedef __attribute__((ext_vector_type(8))) float        v8f;
typedef __attribute__((ext_vector_type(4))) unsigned int v4u;
typedef __attribute__((ext_vector_type(8))) int          v8i;
typedef __attribute__((ext_vector_type(4))) int          v4i;

// ---------------------------------------------------------------------------
// Kernel 1: sparse softmax over the K=10 edges of each row, per head.
// Produces alpha (with self-edge zeroed, matching A_off) and diag = A[i,i].
// Exactly matches the dense reference: exp(NEG - max) underflows to 0 in f32.
// ---------------------------------------------------------------------------
__global__ __launch_bounds__(128)
void semgconv_prep(const float* __restrict__ e, const int* __restrict__ cols,
                   float* __restrict__ alphaW, float* __restrict__ diagW) {
    const int h = blockIdx.x;
    const int i = threadIdx.x;
    if (i >= NB) return;
    const float* er = e + h * (NB * KNZ) + i * KNZ;
    float m = er[0];
    #pragma unroll
    for (int k = 1; k < KNZ; ++k) m = fmaxf(m, er[k]);
    float ex[KNZ];
    float s = 0.f;
    #pragma unroll
    for (int k = 0; k < KNZ; ++k) { ex[k] = expf(er[k] - m); s += ex[k]; }
    const float inv = 1.f / s;
    float dg = 0.f;
    #pragma unroll
    for (int k = 0; k < KNZ; ++k) {
        float a = ex[k] * inv;
        if (cols[i * KNZ + k] == i) { dg += a; a = 0.f; }  // diagonal -> diag, zero in A_off
        alphaW[(h * NB + i) * KNZ + k] = a;
    }
    diagW[h * NB + i] = dg;
}

// ---------------------------------------------------------------------------
// 7 M-tile (112 rows) x 16-col GEMM slab over K=256 using V_WMMA_F32_16X16X4_F32.
// A frags from LDS (x staged, stride 260 -> all-64-bank b64 reads),
// B frags strided from global W (L2 resident, 512KB total).
// A layout (16x4 f32): lane m=l&15, VGPR v -> K = ks + v + 2*(l>>4)
// B layout (4x16 f32): lane n=l&15, VGPR v -> K = ks + v + 2*(l>>4)
// D layout: VGPR v -> row M = v + 8*(l>>4), col N = l&15
// ---------------------------------------------------------------------------
__device__ __forceinline__
void gemm7(const float* __restrict__ xl, const float* __restrict__ wb,
           const int l15, const int khalf, v8f acc[7]) {
    const v8f z = {0.f, 0.f, 0.f, 0.f, 0.f, 0.f, 0.f, 0.f};
    #pragma unroll
    for (int mt = 0; mt < 7; ++mt) acc[mt] = z;
    #pragma unroll 2
    for (int ks = 0; ks < CIN; ks += 4) {
        const int k0 = ks + 2 * khalf;
        v2f bf;
        bf.x = wb[(size_t)k0 * HC];
        bf.y = wb[(size_t)(k0 + 1) * HC];
        #pragma unroll
        for (int mt = 0; mt < 7; ++mt) {
            const v2f af = *(const v2f*)(&xl[(mt * 16 + l15) * XS + k0]);
            acc[mt] = __builtin_amdgcn_wmma_f32_16x16x4_f32(
                false, af, false, bf, (short)0, acc[mt], false, false);
        }
    }
}

// ---------------------------------------------------------------------------
// Kernel 2: fully fused per-batch-element kernel. One block per b (1024 blocks,
// 256 threads = 8 waves). x[b] is DMA'd into LDS by the Tensor Data Mover with
// hardware row padding (256 dwords + 4 pad = stride 260). Phase 1: h1 = x@W[:,1]
// into LDS. Phase 2: h0 tiles stay in registers; sparse-aggregate neighbors
// from h1 LDS; write out.
// ---------------------------------------------------------------------------
__global__ __launch_bounds__(256)
void semgconv_main(const float* __restrict__ x, const float* __restrict__ W,
                   const float* __restrict__ alphaW, const float* __restrict__ diagW,
                   const int* __restrict__ cols, const float* __restrict__ bias,
                   float* __restrict__ out) {
    __shared__ float x_lds[MPAD * XS];        // 116,480 B (TDM fills rows 0..97)
    __shared__ float h1_lds[NB * H1S];        // 101,920 B
    __shared__ float s_alpha[NH * NB * KNZ];  //  15,680 B
    __shared__ float s_diag[NH * NB];         //   1,568 B
    __shared__ int   s_cols[NB * KNZ];        //   3,920 B   => ~234 KB of 320 KB WGP LDS

    const int tid   = threadIdx.x;
    const int b     = blockIdx.x;
    const int w     = tid >> 5;    // wave 0..7
    const int lane  = tid & 31;
    const int l15   = lane & 15;
    const int khalf = lane >> 4;

    const float* xb = x + (size_t)b * (NB * CIN);

    // Zero the 14 pad rows (98..111) that the TDM does not write.
    for (int idx = tid; idx < (MPAD - NB) * XS; idx += 256) x_lds[NB * XS + idx] = 0.f;
    // Stage the small softmax/topology tables.
    for (int idx = tid; idx < NB * KNZ; idx += 256)      s_cols[idx]  = cols[idx];
    for (int idx = tid; idx < NH * NB * KNZ; idx += 256) s_alpha[idx] = alphaW[idx];
    for (int idx = tid; idx < NH * NB; idx += 256)       s_diag[idx]  = diagW[idx];

#if __has_builtin(__builtin_amdgcn_tensor_load_to_lds)
    // --- Tensor Data Mover: DMA the 98x256 f32 tile with hardware padding ---
    // pad_interval code 7 = 256 dwords, pad_amount code 3 = 4 dwords -> LDS row
    // stride 260 floats, matching the bank-conflict-free A-fragment layout.
    if (w == 0) {   // TDM ignores EXEC: exactly one wave must issue the descriptor
        const unsigned           lds_base = (unsigned)(unsigned long long)(uintptr_t)&x_lds[0];
        const unsigned long long ga       = (unsigned long long)(uintptr_t)xb;
        const v4u g0 = { 1u,                                   // count=1 valid user D#
                         lds_base,                             // lds_addr (bytes)
                         (unsigned)ga,                         // global_addr[31:0]
                         (unsigned)((ga >> 32) & 0x01FFFFFFu)  // global_addr[56:32]
                             | (2u << 30) };                   // type=2 (image)
        const v8i g1 = { (int)((2u << 16)        // data_size = 4B
                             | (1u << 20)        // pad_enable
                             | (7u << 22)        // pad_interval: 256 dwords
                             | (3u << 25)),      // pad_amount:   4 dwords
                         (int)(256u << 16),      // tensor_dim0 = 256 (bits 79:48)
                         (int)(98u  << 16),      // tensor_dim1 = 98  (bits 111:80)
                         (int)(256u << 16),      // tile_dim0   = 256 (bits 127:112)
                         (int)98u,               // tile_dim1   = 98  (bits 143:128)
                         (int)256u,              // tensor_dim0_stride = 256 (bits 207:160)
                         0, 0 };
        const v4i zz4 = { 0, 0, 0, 0 };          // groups 2/3 unused (2D tile)
        const v8i zz8 = { 0, 0, 0, 0, 0, 0, 0, 0 };
        __builtin_amdgcn_tensor_load_to_lds(g0, g1, zz4, zz4, zz8, 0);
        __builtin_amdgcn_s_wait_tensorcnt((short)0);
    }
#else
    // Fallback: cooperative vector-load staging of x[b] into padded LDS.
    for (int idx = tid; idx < NB * (CIN / 4); idx += 256) {
        const int r  = idx >> 6;           // CIN/4 = 64 float4 per row
        const int c4 = (idx & 63) << 2;
        float4 v = *(const float4*)(xb + r * CIN + c4);
        *(float4*)(&x_lds[r * XS + c4]) = v;   // byte off = 1040*r + 4*c4, 16B aligned
    }
#endif
    __syncthreads();

    // ---- Phase 1: h1 tiles (col tiles 16..31 of the fused [256,512] W) ----
    #pragma unroll
    for (int t = 0; t < 2; ++t) {
        const int ct = 16 + w + 8 * t;            // this wave's h1 col tile
        const int h  = (ct >> 2) & 3;
        const int d0 = (ct & 3) * 16;
        const float* wb = W + ((size_t)(h * 2 + 1)) * (CIN * HC) + d0 + l15;  // s=1 slice
        v8f acc[7];
        gemm7(x_lds, wb, l15, khalf, acc);
        const int cc = (ct - 16) * 16 + l15;      // h1 channel = h*64+d, 0..255
        #pragma unroll
        for (int mt = 0; mt < 7; ++mt) {
            #pragma unroll
            for (int v = 0; v < 8; ++v) {
                const int r = mt * 16 + khalf * 8 + v;
                if (r < NB) h1_lds[r * H1S + cc] = acc[mt][v];
            }
        }
    }
    __syncthreads();  // all of h1[b] resident in LDS

    // ---- Phase 2: h0 tiles (col tiles 0..15) + sparse aggregation ----
    #pragma unroll
    for (int t = 0; t < 2; ++t) {
        const int ct = w + 8 * t;
        const int h  = ct >> 2;
        const int d0 = (ct & 3) * 16;
        const float* wb = W + ((size_t)(h * 2 + 0)) * (CIN * HC) + d0 + l15;  // s=0 slice
        v8f acc[7];
        gemm7(x_lds, wb, l15, khalf, acc);
        const int cc = ct * 16 + l15;             // output channel = h*64+d
        const float bs = bias[cc];
        #pragma unroll
        for (int mt = 0; mt < 7; ++mt) {
            #pragma unroll
            for (int v = 0; v < 8; ++v) {
                const int i = mt * 16 + khalf * 8 + v;
                if (i < NB) {
                    float val = s_diag[h * NB + i] * acc[mt][v] + bs;
                    const int eb = i * KNZ;
                    #pragma unroll
                    for (int k = 0; k < KNZ; ++k) {
                        const int j = s_cols[eb + k];
                        val += s_alpha[h * NB * KNZ + eb + k] * h1_lds[j * H1S + cc];
                    }
                    out[((size_t)b * NB + i) * COUT + cc] = val;
                }
            }
        }
    }
}

// ---------------------------------------------------------------------------
extern "C" void kernel_launch(void* const* d_in, const int* in_sizes, int n_in,
                              void* d_out, int out_size, void* d_ws, size_t ws_size,
                              hipStream_t stream) {
    (void)in_sizes; (void)n_in; (void)out_size; (void)ws_size;
    const float* x    = (const float*)d_in[0];
    const float* W    = (const float*)d_in[1];
    const float* e    = (const float*)d_in[2];
    const float* bias = (const float*)d_in[3];
    const int*   cols = (const int*)d_in[5];   // rows (d_in[4]) implied by layout

    float* alphaW = (float*)d_ws;                      // [NH*NB*KNZ]
    float* diagW  = alphaW + NH * NB * KNZ;            // [NH*NB]

    semgconv_prep<<<dim3(NH), dim3(128), 0, stream>>>(e, cols, alphaW, diagW);
    semgconv_main<<<dim3(1024), dim3(256), 0, stream>>>(x, W, alphaW, diagW, cols,
                                                        bias, (float*)d_out);
}